// DiffusionBlock_52441550684207
// MI455X (gfx1250) — compile-verified
//
#include <hip/hip_runtime.h>
#include <math.h>
#include <stdint.h>

#define NTOT 65536
#define CH   256
#define KE   128
#define NG   32
#define NPG  2048

typedef float v2f __attribute__((ext_vector_type(2)));
typedef float v8f __attribute__((ext_vector_type(8)));

// D = A(16x4) * B(4x16) + C(16x16), fp32, wave32.
#define WMMA4(a, b, c) \
  __builtin_amdgcn_wmma_f32_16x16x4_f32(false, (a), false, (b), (short)0, (c), false, false)

// CDNA5 direct global->LDS async copy, 16B per lane, tracked by ASYNCcnt.
// VDST operand = LDS byte offset (low 32 bits of the generic shared pointer,
// per ISA 10.2 aperture mapping), VADDR = 64-bit global address.
__device__ __forceinline__ void async_ld_lds16(void* lds_dst, const void* gsrc)
{
  asm volatile("global_load_async_to_lds_b128 %0, %1, off"
               :
               : "v"((uint32_t)(uintptr_t)lds_dst), "v"(gsrc)
               : "memory");
}
__device__ __forceinline__ void wait_async0()
{
  asm volatile("s_wait_asynccnt 0x0" ::: "memory");
}

// ---------------------------------------------------------------------------
// Weight transpose: in[rows][cols] -> out[cols][rows] so WMMA B-operand reads
// B[k][n] with n contiguous (coalesced 64B per 16-lane group).
// ---------------------------------------------------------------------------
__global__ __launch_bounds__(256) void k_transpose(
    const float* __restrict__ in, float* __restrict__ outp, int rows, int cols)
{
  const int idx = blockIdx.x * 256 + threadIdx.x;
  if (idx < rows * cols) {
    const int r = idx / cols;
    const int c = idx % cols;
    outp[c * rows + r] = in[r * cols + c];
  }
}

// ---------------------------------------------------------------------------
// COO segment-sum: gf[row] += vals * x[col].  One wave per edge, 8 coalesced
// dwords per lane, hardware f32 atomic add (relaxed / agent scope).
// ---------------------------------------------------------------------------
__global__ __launch_bounds__(256) void k_edge_scatter(
    const float* __restrict__ x, const int* __restrict__ row,
    const int* __restrict__ col, const float* __restrict__ vals,
    float* __restrict__ gf, int ne)
{
  const int lane = threadIdx.x & 31;
  const int wave = (blockIdx.x * blockDim.x + threadIdx.x) >> 5;
  const int nw   = (gridDim.x * blockDim.x) >> 5;
  for (int e = wave; e < ne; e += nw) {
    const int   r = row[e];
    const int   c = col[e];
    const float v = vals[e];
    const float* xr = x + (long)c * CH;
    float*       gr = gf + (long)r * CH;
#pragma unroll
    for (int j = 0; j < 8; ++j) {
      (void)__hip_atomic_fetch_add(&gr[lane + 32 * j], v * xr[lane + 32 * j],
                                   __ATOMIC_RELAXED, __HIP_MEMORY_SCOPE_AGENT);
    }
  }
}

// ---------------------------------------------------------------------------
// Stage 1: S[g,k,c] = sum_n E[g,n,k] * x[g,n,c] * m[g,n], then fuse the
// spectral decay: Sd = S * exp(-|t_c| * max(lambda,0)).
// ---------------------------------------------------------------------------
__global__ __launch_bounds__(256) void k_spectral_project(
    const float* __restrict__ x, const float* __restrict__ evecs,
    const float* __restrict__ mass, const float* __restrict__ evals,
    const float* __restrict__ tparams, float* __restrict__ Sd)
{
  const int g    = blockIdx.x >> 2;
  const int c0   = (blockIdx.x & 3) * 64;
  const int wave = threadIdx.x >> 5;
  const int lane = threadIdx.x & 31;
  const int l15  = lane & 15;
  const int kh   = lane >> 4;
  const int m0   = wave * 16;
  const long nb  = (long)g * NPG;

  v8f acc[4] = {};
  for (int r0 = 0; r0 < NPG; r0 += 4) {
    const int r = r0 + 2 * kh;
    v2f a;
    a.x = evecs[(nb + r    ) * KE + m0 + l15];
    a.y = evecs[(nb + r + 1) * KE + m0 + l15];
    const float mw0 = mass[nb + r];
    const float mw1 = mass[nb + r + 1];
#pragma unroll
    for (int nt = 0; nt < 4; ++nt) {
      const int c = c0 + nt * 16 + l15;
      v2f b;
      b.x = x[(nb + r    ) * CH + c] * mw0;
      b.y = x[(nb + r + 1) * CH + c] * mw1;
      acc[nt] = WMMA4(a, b, acc[nt]);
    }
  }
#pragma unroll
  for (int nt = 0; nt < 4; ++nt) {
    const int   c  = c0 + nt * 16 + l15;
    const float tc = fabsf(tparams[c]);
#pragma unroll
    for (int r8 = 0; r8 < 8; ++r8) {
      const int kk = m0 + kh * 8 + r8;
      float ev = evals[g * KE + kk];
      ev = ev > 0.f ? ev : 0.f;
      Sd[((long)g * KE + kk) * CH + c] = acc[nt][r8] * __expf(-tc * ev);
    }
  }
}

// ---------------------------------------------------------------------------
// Stage 2: diff[g,n,c] = sum_k E[g,n,k] * Sd[g,k,c].
// E strip (64 nodes x 128) staged via GLOBAL_LOAD_ASYNC_TO_LDS_B128,
// pitch KE+4 -> bank = 4m+k pattern, conflict-free WMMA A reads.
// ---------------------------------------------------------------------------
__global__ __launch_bounds__(256) void k_spectral_expand(
    const float* __restrict__ evecs, const float* __restrict__ Sd,
    float* __restrict__ diff)
{
  const int g    = blockIdx.x >> 5;
  const int mblk = (blockIdx.x & 31) * 64;
  __shared__ float Al[64 * (KE + 4)];
  const int tid = threadIdx.x;
  for (int i = tid; i < 64 * (KE / 4); i += 256) {
    const int rr = i >> 5;
    const int cc = (i & 31) * 4;
    async_ld_lds16(&Al[rr * (KE + 4) + cc],
                   &evecs[((long)g * NPG + mblk + rr) * KE + cc]);
  }
  wait_async0();
  __syncthreads();
  const int wave = tid >> 5, lane = tid & 31, l15 = lane & 15, kh = lane >> 4;
  const int mt  = wave >> 1;
  const int ntb = (wave & 1) * 8;
  v8f acc[8] = {};
  for (int k0 = 0; k0 < KE; k0 += 4) {
    const int k = k0 + 2 * kh;
    v2f a;
    a.x = Al[(mt * 16 + l15) * (KE + 4) + k];
    a.y = Al[(mt * 16 + l15) * (KE + 4) + k + 1];
#pragma unroll
    for (int nt = 0; nt < 8; ++nt) {
      const int c = (ntb + nt) * 16 + l15;
      v2f b;
      b.x = Sd[((long)g * KE + k    ) * CH + c];
      b.y = Sd[((long)g * KE + k + 1) * CH + c];
      acc[nt] = WMMA4(a, b, acc[nt]);
    }
  }
#pragma unroll
  for (int nt = 0; nt < 8; ++nt) {
    const int c = (ntb + nt) * 16 + l15;
#pragma unroll
    for (int r8 = 0; r8 < 8; ++r8) {
      const int node = mblk + mt * 16 + kh * 8 + r8;
      diff[((long)g * NPG + node) * CH + c] = acc[nt][r8];
    }
  }
}

// ---------------------------------------------------------------------------
// Stage 3: gf <- gf @ grad_W^T + grad_b, IN PLACE (strip staged into LDS via
// async copy before the overwrite; no other block touches this strip).
// ---------------------------------------------------------------------------
__global__ __launch_bounds__(256) void k_grad_linear(
    float* __restrict__ gf, const float* __restrict__ gWt,
    const float* __restrict__ gb)
{
  const int m0 = blockIdx.x * 32;
  __shared__ float Al[32 * (CH + 4)];
  const int tid = threadIdx.x;
  for (int i = tid; i < 32 * (CH / 4); i += 256) {
    const int rr = i >> 6;
    const int cc = (i & 63) * 4;
    async_ld_lds16(&Al[rr * (CH + 4) + cc],
                   &gf[((long)(m0 + rr)) * CH + cc]);
  }
  wait_async0();
  __syncthreads();
  const int wave = tid >> 5, lane = tid & 31, l15 = lane & 15, kh = lane >> 4;
  const int mt  = wave & 1;
  const int ntb = (wave >> 1) * 4;
  v8f acc[4] = {};
  for (int k0 = 0; k0 < CH; k0 += 4) {
    const int k = k0 + 2 * kh;
    v2f a;
    a.x = Al[(mt * 16 + l15) * (CH + 4) + k];
    a.y = Al[(mt * 16 + l15) * (CH + 4) + k + 1];
#pragma unroll
    for (int j = 0; j < 4; ++j) {
      const int c = (ntb + j) * 16 + l15;
      v2f b;
      b.x = gWt[(k    ) * CH + c];
      b.y = gWt[(k + 1) * CH + c];
      acc[j] = WMMA4(a, b, acc[j]);
    }
  }
#pragma unroll
  for (int j = 0; j < 4; ++j) {
    const int   c  = (ntb + j) * 16 + l15;
    const float bb = gb[c];
#pragma unroll
    for (int r8 = 0; r8 < 8; ++r8) {
      const int m = mt * 16 + kh * 8 + r8;
      gf[((long)(m0 + m)) * CH + c] = acc[j][r8] + bb;
    }
  }
}

// ---------------------------------------------------------------------------
// Stage 4 megakernel, per 32-row strip:
//   fused = [diff || g2]   (LDS via async copy, pitch 516)
//   h     = relu(fused @ W1^T + b1)           (LDS, pitch 260)
//   y     = h @ W2^T + b2 + x                 (reuses fused LDS buffer)
//   out   = LayerNorm(y) * g + b              (wave-level reduction)
// ---------------------------------------------------------------------------
#define AP (2 * CH + 4) /* 516 */
#define HP (CH + 4)     /* 260 */

__global__ __launch_bounds__(256) void k_mlp_ln(
    const float* __restrict__ x, const float* __restrict__ diff,
    const float* __restrict__ g2,
    const float* __restrict__ W1t, const float* __restrict__ b1,
    const float* __restrict__ W2t, const float* __restrict__ b2,
    const float* __restrict__ lng, const float* __restrict__ lnb,
    float* __restrict__ out)
{
  const int m0 = blockIdx.x * 32;
  __shared__ float sA[32 * AP]; // fused input; later reused as pre-LN y
  __shared__ float sH[32 * HP];
  const int tid = threadIdx.x;
  for (int i = tid; i < 32 * 64; i += 256) {
    const int rr = i >> 6;
    const int cc = (i & 63) * 4;
    async_ld_lds16(&sA[rr * AP + cc],
                   &diff[((long)(m0 + rr)) * CH + cc]);
    async_ld_lds16(&sA[rr * AP + CH + cc],
                   &g2[((long)(m0 + rr)) * CH + cc]);
  }
  wait_async0();
  __syncthreads();
  const int wave = tid >> 5, lane = tid & 31, l15 = lane & 15, kh = lane >> 4;
  const int mt  = wave & 1;
  const int ntb = (wave >> 1) * 4;

  { // h = relu(fused @ W1^T + b1)
    v8f acc[4] = {};
    for (int k0 = 0; k0 < 2 * CH; k0 += 4) {
      const int k = k0 + 2 * kh;
      v2f a;
      a.x = sA[(mt * 16 + l15) * AP + k];
      a.y = sA[(mt * 16 + l15) * AP + k + 1];
#pragma unroll
      for (int j = 0; j < 4; ++j) {
        const int c = (ntb + j) * 16 + l15;
        v2f b;
        b.x = W1t[(k    ) * CH + c];
        b.y = W1t[(k + 1) * CH + c];
        acc[j] = WMMA4(a, b, acc[j]);
      }
    }
#pragma unroll
    for (int j = 0; j < 4; ++j) {
      const int   c  = (ntb + j) * 16 + l15;
      const float bb = b1[c];
#pragma unroll
      for (int r8 = 0; r8 < 8; ++r8) {
        const int   m = mt * 16 + kh * 8 + r8;
        const float v = acc[j][r8] + bb;
        sH[m * HP + c] = v > 0.f ? v : 0.f;
      }
    }
  }
  __syncthreads(); // all waves finished reading sA -> safe to reuse as y

  { // y = h @ W2^T + b2 + x
    v8f acc[4] = {};
    for (int k0 = 0; k0 < CH; k0 += 4) {
      const int k = k0 + 2 * kh;
      v2f a;
      a.x = sH[(mt * 16 + l15) * HP + k];
      a.y = sH[(mt * 16 + l15) * HP + k + 1];
#pragma unroll
      for (int j = 0; j < 4; ++j) {
        const int c = (ntb + j) * 16 + l15;
        v2f b;
        b.x = W2t[(k    ) * CH + c];
        b.y = W2t[(k + 1) * CH + c];
        acc[j] = WMMA4(a, b, acc[j]);
      }
    }
#pragma unroll
    for (int j = 0; j < 4; ++j) {
      const int   c  = (ntb + j) * 16 + l15;
      const float bb = b2[c];
#pragma unroll
      for (int r8 = 0; r8 < 8; ++r8) {
        const int m = mt * 16 + kh * 8 + r8;
        sA[m * AP + c] = acc[j][r8] + bb + x[((long)(m0 + m)) * CH + c];
      }
    }
  }
  __syncthreads();

  // LayerNorm: one wave per 4 rows, conflict-free (bank = lane + 32j + 4r).
#pragma unroll
  for (int rr = 0; rr < 4; ++rr) {
    const int r = wave * 4 + rr;
    float v[8], s = 0.f, ss = 0.f;
#pragma unroll
    for (int j = 0; j < 8; ++j) {
      const float t = sA[r * AP + lane + 32 * j];
      v[j] = t; s += t; ss += t * t;
    }
    for (int off = 16; off > 0; off >>= 1) {
      s  += __shfl_xor(s,  off, 32);
      ss += __shfl_xor(ss, off, 32);
    }
    const float mu  = s * (1.f / CH);
    const float var = ss * (1.f / CH) - mu * mu;
    const float inv = rsqrtf(var + 1e-5f);
#pragma unroll
    for (int j = 0; j < 8; ++j) {
      const int c = lane + 32 * j;
      out[((long)(m0 + r)) * CH + c] = (v[j] - mu) * inv * lng[c] + lnb[c];
    }
  }
}

// ---------------------------------------------------------------------------
extern "C" void kernel_launch(void* const* d_in, const int* in_sizes, int n_in,
                              void* d_out, int out_size, void* d_ws, size_t ws_size,
                              hipStream_t stream)
{
  (void)n_in; (void)out_size; (void)ws_size;
  const float* x     = (const float*)d_in[0];
  const float* evals = (const float*)d_in[1];
  const float* evecs = (const float*)d_in[2];
  const float* mass  = (const float*)d_in[3];
  const int*   row   = (const int*)d_in[4];
  const int*   col   = (const int*)d_in[5];
  const float* vals  = (const float*)d_in[6];
  const float* tpar  = (const float*)d_in[7];
  const float* gW    = (const float*)d_in[8];
  const float* gb    = (const float*)d_in[9];
  const float* W1    = (const float*)d_in[10];
  const float* b1    = (const float*)d_in[11];
  const float* W2    = (const float*)d_in[12];
  const float* b2    = (const float*)d_in[13];
  const float* lng   = (const float*)d_in[14];
  const float* lnb   = (const float*)d_in[15];
  float* out = (float*)d_out;

  float* ws   = (float*)d_ws;
  float* gf   = ws;                                  // [N,C] scatter accum -> g2 (in place)
  float* diff = gf   + (size_t)NTOT * CH;            // [N,C]
  float* Sd   = diff + (size_t)NTOT * CH;            // [G,K,C] decayed spectrum
  float* gWt  = Sd   + (size_t)NG * KE * CH;         // [C,C]
  float* W1t  = gWt  + (size_t)CH * CH;              // [2C,C]
  float* W2t  = W1t  + (size_t)2 * CH * CH;          // [C,C]

  const int ne = in_sizes[4];

  hipMemsetAsync(gf, 0, (size_t)NTOT * CH * sizeof(float), stream);
  k_transpose<<<(CH * CH + 255) / 256, 256, 0, stream>>>(gW, gWt, CH, CH);
  k_transpose<<<(CH * 2 * CH + 255) / 256, 256, 0, stream>>>(W1, W1t, CH, 2 * CH);
  k_transpose<<<(CH * CH + 255) / 256, 256, 0, stream>>>(W2, W2t, CH, CH);
  k_edge_scatter<<<4096, 256, 0, stream>>>(x, row, col, vals, gf, ne);
  k_spectral_project<<<NG * 4, 256, 0, stream>>>(x, evecs, mass, evals, tpar, Sd);
  k_spectral_expand<<<NG * 32, 256, 0, stream>>>(evecs, Sd, diff);
  k_grad_linear<<<NTOT / 32, 256, 0, stream>>>(gf, gWt, gb);
  k_mlp_ln<<<NTOT / 32, 256, 0, stream>>>(x, diff, gf, W1t, b1, W2t, b2, lng, lnb, out);
}